// GCN_63479616635263
// MI455X (gfx1250) — compile-verified
//
#include <hip/hip_runtime.h>
#include <hip/hip_bf16.h>

typedef __attribute__((ext_vector_type(2))) float v2f;
typedef __attribute__((ext_vector_type(8))) float v8f;

// ---------------------------------------------------------------------------
// zero fill (avoid hipMemsetAsync questions under graph capture)
// ---------------------------------------------------------------------------
__global__ __launch_bounds__(256) void zero_kernel(float* __restrict__ p, size_t n) {
    size_t i = (size_t)blockIdx.x * blockDim.x + threadIdx.x;
    if (i < n) p[i] = 0.0f;
}

// ---------------------------------------------------------------------------
// Edge scatter-add: dst[didx[e]][:] += src[sidx[e]][:], F floats per row,
// float4-vectorized: 16 (F=64) or 32 (F=128) threads per edge.
// Node tables fit in 192MB L2 -> atomics and gathers are L2-resident.
// ---------------------------------------------------------------------------
template <int F>
__global__ __launch_bounds__(256) void scatter_add_vec(const float* __restrict__ src,
                                                       const int* __restrict__ sidx,
                                                       const int* __restrict__ didx,
                                                       float* dst, int nE) {
    constexpr int PER = F / 4;
    unsigned t = blockIdx.x * blockDim.x + threadIdx.x;
    unsigned e = t / PER;
    unsigned f = (t % PER) * 4;
    if (e >= (unsigned)nE) return;
    int s = sidx[e];
    int d = didx[e];
    const float4 v = *(const float4*)(src + (size_t)s * F + f);
    float* p = dst + (size_t)d * F + f;
    atomicAdd(p + 0, v.x);
    atomicAdd(p + 1, v.y);
    atomicAdd(p + 2, v.z);
    atomicAdd(p + 3, v.w);
}

// ---------------------------------------------------------------------------
// Dual-input WMMA GEMM (exact fp32 via V_WMMA_F32_16X16X4_F32):
//   out[M x 128] = A1[M x K1] @ W1[K1 x 128] + A2[M x K2] @ W2[K2 x 128] + bias
// One wave32 computes a (16*MT) x 128 strip: B fragments are reused across
// MT m-tiles (B traffic / FLOP cut MT x), A fragments reused 8x.
// Output may alias A1 (rows are wave-private; all A1 loads feed accumulators
// before the epilogue stores).
// ISA layouts (CDNA5 ISA):
//   A 16x4 : lane(0..15)=row, vgpr pair = K {0,1} (lo lanes) / {2,3} (hi lanes)
//   B 4x16 : lane=col, vgpr pair = K {0,1} / {2,3} by lane half
//   C/D    : vgpr i = row m0+i (lanes 0-15) / m0+8+i (lanes 16-31), lane=col
// M must be a multiple of 16 (tail handled at 16-row-tile granularity).
// ---------------------------------------------------------------------------
template <int K1, int K2, bool RELU, int MT>
__global__ __launch_bounds__(256) void gemm_dual_wmma(const float* __restrict__ A1,
                                                      const float* __restrict__ W1,
                                                      const float* __restrict__ A2,
                                                      const float* __restrict__ W2,
                                                      const float* __restrict__ bias,
                                                      float* out, int M) {
    constexpr int N = 128;
    const int lane = threadIdx.x & 31;
    const int wave = threadIdx.x >> 5;
    const int m0 = (blockIdx.x * 8 + wave) * (16 * MT);
    if (m0 >= M) return;
    const int half = lane >> 4;  // 0: K pair {0,1} / rows +0..7 ; 1: {2,3} / +8
    const int lr = lane & 15;

    v8f acc[MT][8] = {};

    int arow[MT];
#pragma unroll
    for (int mt = 0; mt < MT; ++mt) {
        int r = m0 + mt * 16 + lr;
        arow[mt] = (r < M) ? r : (M - 1);  // clamp OOB tail tiles; stores skipped
    }

    for (int k = 0; k < K1; k += 4) {
        v2f a[MT];
#pragma unroll
        for (int mt = 0; mt < MT; ++mt)
            a[mt] = *(const v2f*)(A1 + (size_t)arow[mt] * K1 + k + 2 * half);
        const float* w0 = W1 + (size_t)(k + 2 * half) * N + lr;
#pragma unroll
        for (int nt = 0; nt < 8; ++nt) {
            v2f b;
            b.x = w0[nt * 16];
            b.y = w0[N + nt * 16];
#pragma unroll
            for (int mt = 0; mt < MT; ++mt)
                acc[mt][nt] = __builtin_amdgcn_wmma_f32_16x16x4_f32(
                    false, a[mt], false, b, (short)0, acc[mt][nt], false, false);
        }
    }
    for (int k = 0; k < K2; k += 4) {
        v2f a[MT];
#pragma unroll
        for (int mt = 0; mt < MT; ++mt)
            a[mt] = *(const v2f*)(A2 + (size_t)arow[mt] * K2 + k + 2 * half);
        const float* w0 = W2 + (size_t)(k + 2 * half) * N + lr;
#pragma unroll
        for (int nt = 0; nt < 8; ++nt) {
            v2f b;
            b.x = w0[nt * 16];
            b.y = w0[N + nt * 16];
#pragma unroll
            for (int mt = 0; mt < MT; ++mt)
                acc[mt][nt] = __builtin_amdgcn_wmma_f32_16x16x4_f32(
                    false, a[mt], false, b, (short)0, acc[mt][nt], false, false);
        }
    }

#pragma unroll
    for (int mt = 0; mt < MT; ++mt) {
        if (m0 + mt * 16 >= M) break;  // M % 16 == 0 -> tile-level guard suffices
#pragma unroll
        for (int nt = 0; nt < 8; ++nt) {
            const int col = nt * 16 + lr;
            const float bv = bias[col];
#pragma unroll
            for (int i = 0; i < 8; ++i) {
                float v = acc[mt][nt][i] + bv;
                if (RELU) v = fmaxf(v, 0.0f);
                out[(size_t)(m0 + mt * 16 + i + 8 * half) * N + col] = v;
            }
        }
    }
}

// ---------------------------------------------------------------------------
// Per-node tiny GEMM: out[n][0..2] = X[n][0..127] @ W[128x3] (+ bias).
// One wave32 per node: float4 coalesced row load + shuffle reduction.
// ---------------------------------------------------------------------------
__global__ __launch_bounds__(256) void rowmat3(const float* __restrict__ X,
                                               const float* __restrict__ W,
                                               const float* __restrict__ bias,
                                               float* __restrict__ out, int n) {
    const int lane = threadIdx.x & 31;
    const int node = blockIdx.x * (blockDim.x >> 5) + (threadIdx.x >> 5);
    if (node >= n) return;
    const float4 x4 = *(const float4*)(X + (size_t)node * 128 + lane * 4);
    const float* w = W + lane * 4 * 3;
    float p0 = x4.x * w[0] + x4.y * w[3] + x4.z * w[6] + x4.w * w[9];
    float p1 = x4.x * w[1] + x4.y * w[4] + x4.z * w[7] + x4.w * w[10];
    float p2 = x4.x * w[2] + x4.y * w[5] + x4.z * w[8] + x4.w * w[11];
#pragma unroll
    for (int off = 16; off; off >>= 1) {
        p0 += __shfl_xor(p0, off, 32);
        p1 += __shfl_xor(p1, off, 32);
        p2 += __shfl_xor(p2, off, 32);
    }
    if (lane == 0) {
        float b0 = 0.f, b1 = 0.f, b2 = 0.f;
        if (bias) { b0 = bias[0]; b1 = bias[1]; b2 = bias[2]; }
        out[(size_t)node * 3 + 0] = p0 + b0;
        out[(size_t)node * 3 + 1] = p1 + b1;
        out[(size_t)node * 3 + 2] = p2 + b2;
    }
}

// 3-feature edge scatter-add (layer 3, post-projection)
__global__ __launch_bounds__(256) void scatter_add3(const float* __restrict__ proj,
                                                    const int* __restrict__ sidx,
                                                    const int* __restrict__ didx,
                                                    float* out, int nE) {
    int e = blockIdx.x * blockDim.x + threadIdx.x;
    if (e >= nE) return;
    int s = sidx[e];
    int d = didx[e];
    atomicAdd(out + (size_t)d * 3 + 0, proj[(size_t)s * 3 + 0]);
    atomicAdd(out + (size_t)d * 3 + 1, proj[(size_t)s * 3 + 1]);
    atomicAdd(out + (size_t)d * 3 + 2, proj[(size_t)s * 3 + 2]);
}

// _add_knowledge: row l2norm*10 then subtract 10*flag on cols 0 and 2
__global__ __launch_bounds__(256) void knowledge_kernel(float* __restrict__ out,
                                                        const float* __restrict__ x,
                                                        int n) {
    int i = blockIdx.x * blockDim.x + threadIdx.x;
    if (i >= n) return;
    float a = out[(size_t)i * 3 + 0];
    float b = out[(size_t)i * 3 + 1];
    float c = out[(size_t)i * 3 + 2];
    float nrm = sqrtf(a * a + b * b + c * c);
    float s = 10.0f / fmaxf(nrm, 1e-12f);
    a *= s; b *= s; c *= s;
    if (x[(size_t)i * 64 + 61] != 0.0f) a -= 10.0f;
    if (x[(size_t)i * 64 + 63] != 0.0f) c -= 10.0f;
    out[(size_t)i * 3 + 0] = a;
    out[(size_t)i * 3 + 1] = b;
    out[(size_t)i * 3 + 2] = c;
}

static inline int ceil_div(long long a, long long b) { return (int)((a + b - 1) / b); }

extern "C" void kernel_launch(void* const* d_in, const int* in_sizes, int n_in,
                              void* d_out, int out_size, void* d_ws, size_t ws_size,
                              hipStream_t stream) {
    const float* x_s = (const float*)d_in[0];
    const float* x_t = (const float*)d_in[1];
    const int* e_l = (const int*)d_in[2];
    const int* e_r = (const int*)d_in[3];
    const float* w1lr_rel  = (const float*)d_in[4];
    const float* b1lr      = (const float*)d_in[5];
    const float* w1lr_root = (const float*)d_in[6];
    const float* w1rl_rel  = (const float*)d_in[7];
    const float* b1rl      = (const float*)d_in[8];
    const float* w1rl_root = (const float*)d_in[9];
    const float* w2lr_rel  = (const float*)d_in[10];
    const float* b2lr      = (const float*)d_in[11];
    const float* w2lr_root = (const float*)d_in[12];
    const float* w2rl_rel  = (const float*)d_in[13];
    const float* b2rl      = (const float*)d_in[14];
    const float* w2rl_root = (const float*)d_in[15];
    const float* w3lr_rel  = (const float*)d_in[16];
    const float* b3lr      = (const float*)d_in[17];
    const float* w3lr_root = (const float*)d_in[18];
    const float* w3rl_rel  = (const float*)d_in[19];
    const float* b3rl      = (const float*)d_in[20];
    const float* w3rl_root = (const float*)d_in[21];

    const int NL = in_sizes[0] / 64;
    const int NR = in_sizes[1] / 64;
    const int E  = in_sizes[2];

    float* ws = (float*)d_ws;
    // workspace layout (floats); layer2 GEMM writes in-place over AGG regions
    const size_t AGG_R = 0;                         // NR*128
    const size_t AGG_L = AGG_R + (size_t)NR * 128;  // NL*128
    const size_t H1_R  = AGG_L + (size_t)NL * 128;  // NR*128
    const size_t H1_L  = H1_R + (size_t)NR * 128;   // NL*128
    // projections reuse the (then-dead) H1_R region
    const size_t PROJ_LR = H1_R;                    // NL*3
    const size_t PROJ_RL = H1_R + (size_t)NL * 3;   // NR*3

    float* d_left  = (float*)d_out;                 // NL*3
    float* d_right = (float*)d_out + (size_t)NL * 3;// NR*3

    const int TPB = 256;
    constexpr int MT = 2;  // 32 output rows per wave -> B traffic halved

    // ---- Layer 1: aggregate 64-feature inputs, then dual GEMM to H=128 ----
    zero_kernel<<<ceil_div((size_t)NR * 64, TPB), TPB, 0, stream>>>(ws + AGG_R, (size_t)NR * 64);
    zero_kernel<<<ceil_div((size_t)NL * 64, TPB), TPB, 0, stream>>>(ws + AGG_L, (size_t)NL * 64);
    scatter_add_vec<64><<<ceil_div((long long)E * 16, TPB), TPB, 0, stream>>>(x_s, e_l, e_r, ws + AGG_R, E);
    scatter_add_vec<64><<<ceil_div((long long)E * 16, TPB), TPB, 0, stream>>>(x_t, e_r, e_l, ws + AGG_L, E);
    gemm_dual_wmma<64, 64, true, MT><<<ceil_div(ceil_div(NR, 16 * MT), 8), TPB, 0, stream>>>(
        ws + AGG_R, w1lr_rel, x_t, w1lr_root, b1lr, ws + H1_R, NR);
    gemm_dual_wmma<64, 64, true, MT><<<ceil_div(ceil_div(NL, 16 * MT), 8), TPB, 0, stream>>>(
        ws + AGG_L, w1rl_rel, x_s, w1rl_root, b1rl, ws + H1_L, NL);

    // ---- Layer 2: aggregate H=128 activations, dual GEMM (output aliases AGG) ----
    zero_kernel<<<ceil_div((size_t)NR * 128, TPB), TPB, 0, stream>>>(ws + AGG_R, (size_t)NR * 128);
    zero_kernel<<<ceil_div((size_t)NL * 128, TPB), TPB, 0, stream>>>(ws + AGG_L, (size_t)NL * 128);
    scatter_add_vec<128><<<ceil_div((long long)E * 32, TPB), TPB, 0, stream>>>(ws + H1_L, e_l, e_r, ws + AGG_R, E);
    scatter_add_vec<128><<<ceil_div((long long)E * 32, TPB), TPB, 0, stream>>>(ws + H1_R, e_r, e_l, ws + AGG_L, E);
    gemm_dual_wmma<128, 128, true, MT><<<ceil_div(ceil_div(NR, 16 * MT), 8), TPB, 0, stream>>>(
        ws + AGG_R, w2lr_rel, ws + H1_R, w2lr_root, b2lr, ws + AGG_R, NR);
    gemm_dual_wmma<128, 128, true, MT><<<ceil_div(ceil_div(NL, 16 * MT), 8), TPB, 0, stream>>>(
        ws + AGG_L, w2rl_rel, ws + H1_L, w2rl_root, b2rl, ws + AGG_L, NL);

    // ---- Layer 3 (project-then-aggregate: 42x less edge traffic) ----
    // proj_src = h2 @ w_rel (NL/NR x 3)
    rowmat3<<<ceil_div(NL, 8), TPB, 0, stream>>>(ws + AGG_L, w3lr_rel, nullptr, ws + PROJ_LR, NL);
    rowmat3<<<ceil_div(NR, 8), TPB, 0, stream>>>(ws + AGG_R, w3rl_rel, nullptr, ws + PROJ_RL, NR);
    // init out = bias + h2_dst @ w_root, then atomically add edge aggregates
    rowmat3<<<ceil_div(NR, 8), TPB, 0, stream>>>(ws + AGG_R, w3lr_root, b3lr, d_right, NR);
    rowmat3<<<ceil_div(NL, 8), TPB, 0, stream>>>(ws + AGG_L, w3rl_root, b3rl, d_left, NL);
    scatter_add3<<<ceil_div(E, TPB), TPB, 0, stream>>>(ws + PROJ_LR, e_l, e_r, d_right, E);
    scatter_add3<<<ceil_div(E, TPB), TPB, 0, stream>>>(ws + PROJ_RL, e_r, e_l, d_left, E);

    // ---- add_knowledge ----
    knowledge_kernel<<<ceil_div(NL, TPB), TPB, 0, stream>>>(d_left, x_s, NL);
    knowledge_kernel<<<ceil_div(NR, TPB), TPB, 0, stream>>>(d_right, x_t, NR);
}